// BahdanauAttention_55963423867293
// MI455X (gfx1250) — compile-verified
//
#include <hip/hip_runtime.h>
#include <math.h>

typedef __attribute__((ext_vector_type(2))) float v2f;
typedef __attribute__((ext_vector_type(8))) float v8f;

// Problem sizes (fixed by reference)
static constexpr int kB = 64;    // batch
static constexpr int kS = 2048;  // source length
static constexpr int kH = 512;   // hidden
static constexpr int kK = 1024;  // encoder dim

// Step-4 chunking
static constexpr int kNCH = 16;           // S chunks
static constexpr int kSCH = kS / kNCH;    // 128 source rows per chunk

// ws layout (in floats)
static constexpr size_t WS_QOUT   = 0;                       // 64*512
static constexpr size_t WS_SCORES = WS_QOUT + (size_t)kB*kH; // 64*2048
static constexpr size_t WS_PART   = WS_SCORES + (size_t)kB*kS; // 64*16*1024

// ---------------------------------------------------------------------------
// Kernel 1: q[b,h] = sum_e query[b,e] * W_q[h,e]   (M=64, N=512, K=512 GEMM)
// One wave per 16x16 output tile, v_wmma_f32_16x16x4_f32 over K.
// A-frag (16x4 f32): lane L holds A[M=L&15][K = 2*(L>>4) + {0,1}]
// B-frag (4x16 f32): lane L holds B[K = 2*(L>>4) + {0,1}][N = L&15]
//   with B[k][n] = W_q[n*kH + k]  -> consecutive floats in a W_q row.
// D (16x16 f32): VGPR r, lanes 0-15 -> M=r, lanes 16-31 -> M=8+r.
// ---------------------------------------------------------------------------
__global__ void qproj_wmma_kernel(const float* __restrict__ query,
                                  const float* __restrict__ Wq,
                                  float* __restrict__ qout) {
    const int wave = (int)((blockIdx.x * blockDim.x + threadIdx.x) >> 5); // 0..127
    const int lane = (int)(threadIdx.x & 31);
    const int mt   = wave & 3;    // 4 tiles over M=64
    const int nt   = wave >> 2;   // 32 tiles over N=512
    const int half = lane >> 4;   // K-pair selector
    const int l15  = lane & 15;

    const float* arow = query + (size_t)(mt * 16 + l15) * kH + half * 2;
    const float* brow = Wq    + (size_t)(nt * 16 + l15) * kH + half * 2;

    v8f acc = {};
    for (int k = 0; k < kH; k += 4) {
        v2f a = *(const v2f*)(arow + k);
        v2f b = *(const v2f*)(brow + k);
        acc = __builtin_amdgcn_wmma_f32_16x16x4_f32(
            /*neg_a=*/false, a, /*neg_b=*/false, b,
            /*c_mod=*/(short)0, acc, /*reuse_a=*/false, /*reuse_b=*/false);
    }

    const int nn = nt * 16 + l15;
#pragma unroll
    for (int r = 0; r < 8; ++r) {
        const int mm = mt * 16 + half * 8 + r;
        qout[(size_t)mm * kH + nn] = acc[r];
    }
}

// ---------------------------------------------------------------------------
// Kernel 2: scores[b,s] = sum_h tanh(q[b,h] + proj_key[b,s,h]) * v_energy[h]
// One wave per score; each lane covers 16 consecutive h (4x float4 loads),
// then a wave32 xor-shuffle reduction. Streams 256 MB of proj_key.
// ---------------------------------------------------------------------------
__global__ void scores_kernel(const float* __restrict__ proj_key,
                              const float* __restrict__ qout,
                              const float* __restrict__ v_energy,
                              float* __restrict__ scores) {
    const int id   = (int)(blockIdx.x * 8 + (threadIdx.x >> 5)); // score index
    const int lane = (int)(threadIdx.x & 31);
    const int b = id >> 11;      // / 2048
    const int s = id & (kS - 1); // % 2048

    const float* pk = proj_key + ((size_t)b * kS + s) * kH + lane * 16;
    const float* q  = qout + (size_t)b * kH + lane * 16;
    const float* ve = v_energy + lane * 16;

    float acc = 0.f;
#pragma unroll
    for (int j = 0; j < 4; ++j) {
        float4 p  = *(const float4*)(pk + j * 4);
        float4 qq = *(const float4*)(q  + j * 4);
        float4 vv = *(const float4*)(ve + j * 4);
        acc += tanhf(p.x + qq.x) * vv.x;
        acc += tanhf(p.y + qq.y) * vv.y;
        acc += tanhf(p.z + qq.z) * vv.z;
        acc += tanhf(p.w + qq.w) * vv.w;
    }
#pragma unroll
    for (int off = 16; off > 0; off >>= 1)
        acc += __shfl_xor(acc, off, 32);
    if (lane == 0) scores[id] = acc;
}

// ---------------------------------------------------------------------------
// Kernel 3: masked softmax over S per batch. One 256-thread block per batch,
// 8 elements per thread, LDS tree reductions for max and sum.
// ---------------------------------------------------------------------------
__global__ void softmax_kernel(const float* __restrict__ scores,
                               const int* __restrict__ mask,
                               float* __restrict__ alphas) {
    __shared__ float red[256];
    const int b = (int)blockIdx.x;
    const int tid = (int)threadIdx.x;

    float v[8];
    float lmax = -INFINITY;
#pragma unroll
    for (int j = 0; j < 8; ++j) {
        const int s = tid + j * 256;
        float x = scores[(size_t)b * kS + s];
        if (mask[(size_t)b * kS + s] == 0) x = -INFINITY;
        v[j] = x;
        lmax = fmaxf(lmax, x);
    }
    red[tid] = lmax;
    __syncthreads();
    for (int st = 128; st > 0; st >>= 1) {
        if (tid < st) red[tid] = fmaxf(red[tid], red[tid + st]);
        __syncthreads();
    }
    const float mx = red[0];
    __syncthreads();

    float lsum = 0.f;
#pragma unroll
    for (int j = 0; j < 8; ++j) {
        v[j] = expf(v[j] - mx);
        lsum += v[j];
    }
    red[tid] = lsum;
    __syncthreads();
    for (int st = 128; st > 0; st >>= 1) {
        if (tid < st) red[tid] += red[tid + st];
        __syncthreads();
    }
    const float inv = 1.0f / red[0];
#pragma unroll
    for (int j = 0; j < 8; ++j)
        alphas[(size_t)b * kS + tid + j * 256] = v[j] * inv;
}

// ---------------------------------------------------------------------------
// Kernel 4: partial context. Block = (chunk c, batch b); alphas chunk cached
// in LDS; each thread owns 4 consecutive k and streams 128 rows of
// encoder_hidden with fully coalesced float4 loads. 1024 blocks in flight.
// ---------------------------------------------------------------------------
__global__ void context_partial_kernel(const float* __restrict__ eh,
                                       const float* __restrict__ alphas,
                                       float* __restrict__ partials) {
    __shared__ float a_sh[kSCH];
    const int c = (int)blockIdx.x;   // 0..15
    const int b = (int)blockIdx.y;   // 0..63
    const int tid = (int)threadIdx.x;

    if (tid < kSCH) a_sh[tid] = alphas[(size_t)b * kS + c * kSCH + tid];
    __syncthreads();

    const int k = tid * 4;
    const float* base = eh + ((size_t)b * kS + (size_t)c * kSCH) * kK + k;
    float4 acc = {0.f, 0.f, 0.f, 0.f};
    for (int s = 0; s < kSCH; ++s) {
        float4 e = *(const float4*)(base + (size_t)s * kK);
        const float a = a_sh[s];
        acc.x += a * e.x; acc.y += a * e.y;
        acc.z += a * e.z; acc.w += a * e.w;
    }
    *(float4*)(partials + ((size_t)(b * kNCH + c)) * kK + k) = acc;
}

// ---------------------------------------------------------------------------
// Kernel 5: reduce the 16 S-chunk partials into context[b,k].
// ---------------------------------------------------------------------------
__global__ void context_reduce_kernel(const float* __restrict__ partials,
                                      float* __restrict__ context) {
    const int b = (int)blockIdx.x;
    const int k = (int)threadIdx.x * 4;
    float4 acc = {0.f, 0.f, 0.f, 0.f};
    for (int c = 0; c < kNCH; ++c) {
        float4 p = *(const float4*)(partials + ((size_t)(b * kNCH + c)) * kK + k);
        acc.x += p.x; acc.y += p.y; acc.z += p.z; acc.w += p.w;
    }
    *(float4*)(context + (size_t)b * kK + k) = acc;
}

// ---------------------------------------------------------------------------
extern "C" void kernel_launch(void* const* d_in, const int* in_sizes, int n_in,
                              void* d_out, int out_size, void* d_ws, size_t ws_size,
                              hipStream_t stream) {
    (void)in_sizes; (void)n_in; (void)out_size; (void)ws_size;

    const float* query    = (const float*)d_in[0]; // (64,1,512)
    const float* proj_key = (const float*)d_in[1]; // (64,2048,512)
    const float* eh       = (const float*)d_in[2]; // (64,2048,1024)
    const int*   mask     = (const int*)  d_in[3]; // (64,1,2048)
    const float* Wq       = (const float*)d_in[4]; // (512,512)
    const float* ve       = (const float*)d_in[5]; // (512,)

    float* out     = (float*)d_out;
    float* context = out;                        // 64*1024 floats
    float* alphas  = out + (size_t)kB * kK;      // 64*2048 floats

    float* ws       = (float*)d_ws;
    float* qout     = ws + WS_QOUT;
    float* scores   = ws + WS_SCORES;
    float* partials = ws + WS_PART;

    // 1) q = query @ W_q^T  (WMMA f32 16x16x4)
    qproj_wmma_kernel<<<32, 128, 0, stream>>>(query, Wq, qout);

    // 2) scores[b,s] = <tanh(q+proj_key), v_energy>  (streams 256 MB)
    scores_kernel<<<(kB * kS) / 8, 256, 0, stream>>>(proj_key, qout, ve, scores);

    // 3) masked softmax -> alphas (written straight into d_out)
    softmax_kernel<<<kB, 256, 0, stream>>>(scores, mask, alphas);

    // 4) partial context over 16 S-chunks (streams 512 MB)
    context_partial_kernel<<<dim3(kNCH, kB), 256, 0, stream>>>(eh, alphas, partials);

    // 5) deterministic sequential reduction of partials -> context
    context_reduce_kernel<<<kB, 256, 0, stream>>>(partials, context);
}